// CEMA_62904091017752
// MI455X (gfx1250) — compile-verified
//
#include <hip/hip_runtime.h>

typedef __attribute__((ext_vector_type(2))) float v2f;
typedef __attribute__((ext_vector_type(8))) float v8f;

namespace {

constexpr float kAlpha = 0.99f;
constexpr int kB = 8, kS = 4096, kD = 2048;

constexpr int kSeg = 32;                // independent segments along S
constexpr int kSegLen = kS / kSeg;      // 128 steps per segment
constexpr int kChunks = kSegLen / 16;   // 8 WMMA chunks per segment
constexpr int kDT = kD / 16;            // 128 d-tiles
constexpr int kWavesPerBlock = 4;
constexpr int kThreads = kWavesPerBlock * 32;  // 128 threads = 4 waves
constexpr int kWaves = kSeg * kDT;             // 4096 waves
constexpr int kBlocks = kWaves / kWavesPerBlock;

constexpr float cpow(float a, int n) {
  float r = 1.f;
  for (int i = 0; i < n; ++i) r *= a;
  return r;
}
constexpr float kA8   = cpow(kAlpha, 8);
constexpr float kA16  = cpow(kAlpha, 16);
constexpr float kASeg = cpow(kAlpha, kSegLen);  // alpha^128

// alpha^n for runtime n in [0,15]; branch-free select chain (no scratch array)
__device__ inline float apow_rt(int n) {
  float v = 1.f;
#pragma unroll
  for (int k = 0; k < 15; ++k) v = (k < n) ? v * kAlpha : v;
  return v;
}

// broadcast value held by lane (col+16) to all lanes (wave32)
__device__ inline float bcast_upper(float v, int col) {
  return __builtin_bit_cast(
      float, __builtin_amdgcn_ds_bpermute((col + 16) << 2,
                                          __builtin_bit_cast(int, v)));
}

// Build the chunk-invariant A operand: lower-triangular EMA kernel
// L[i][j] = (1-alpha)*alpha^(i-j) for j<=i. A_k = L[:, 4k:4k+4] (16x4 f32).
// f32 A layout: lane m in 0..15 -> row m; VGPR0/1 hold K = 2*half + {0,1}.
__device__ inline void build_A(v2f A[4], int col, int half) {
#pragma unroll
  for (int k = 0; k < 4; ++k) {
    const int j0 = 4 * k + 2 * half;
    A[k].x = (col >= j0)     ? (1.f - kAlpha) * apow_rt(col - j0)     : 0.f;
    A[k].y = (col >= j0 + 1) ? (1.f - kAlpha) * apow_rt(col - j0 - 1) : 0.f;
  }
}

}  // namespace

// Phase 1: per (segment p, d-tile t) wave: compute batch mean m (staged to
// out[0] slice), run local EMA scan (zero initial carry) via WMMA to obtain
// the segment-final tail E_p[d] -> d_ws.
__global__ __launch_bounds__(kThreads) void cema_partial(
    const float* __restrict__ x, float* __restrict__ m_out,
    float* __restrict__ seg_end) {
  const int wave = blockIdx.x * kWavesPerBlock + (threadIdx.x >> 5);
  const int p = wave / kDT;
  const int t = wave - p * kDT;
  const int lane = threadIdx.x & 31;
  const int half = lane >> 4;
  const int col = lane & 15;
  const int d0 = t * 16;
  const int sseg = p * kSegLen;

  v2f A[4];
  build_A(A, col, half);

  float carry = 0.f;
  for (int c = 0; c < kChunks; ++c) {
    const int s0 = sseg + c * 16;
    if (c + 1 < kChunks) {
#pragma unroll
      for (int b = 0; b < kB; ++b)  // prefetch next chunk's leading rows
        __builtin_prefetch(
            x + ((size_t)b * kS + (s0 + 16 + 2 * half)) * kD + d0 + col, 0, 0);
    }
    v8f acc = {};
#pragma unroll
    for (int k = 0; k < 4; ++k) {
      const int r0 = 4 * k + 2 * half;  // f32 B layout: V0/V1 = K rows r0,r0+1
      float sx = 0.f, sy = 0.f;
#pragma unroll
      for (int b = 0; b < kB; ++b) {
        const float* px = x + ((size_t)b * kS + (s0 + r0)) * kD + d0 + col;
        sx += px[0];
        sy += px[kD];
      }
      v2f Bv;
      Bv.x = sx * (1.f / kB);
      Bv.y = sy * (1.f / kB);
      m_out[(size_t)(s0 + r0) * kD + d0 + col] = Bv.x;
      m_out[(size_t)(s0 + r0 + 1) * kD + d0 + col] = Bv.y;
      acc = __builtin_amdgcn_wmma_f32_16x16x4_f32(false, A[k], false, Bv,
                                                  (short)0, acc, false, false);
    }
    // row 15 lives in upper-half lanes, C/D VGPR 7
    const float e15 = acc[7] + kA16 * carry;
    carry = bcast_upper(e15, col);
  }
  if (half == 0) seg_end[(size_t)p * kD + d0 + col] = carry;
}

// Phase 2: combine prior segments' tails into the true incoming carry,
// rescan the segment from staged m via WMMA, add carry term, and write the
// broadcast output for all 8 batch entries.
__global__ __launch_bounds__(kThreads) void cema_apply(
    const float* __restrict__ m_in, const float* __restrict__ seg_end,
    float* __restrict__ out) {
  const int wave = blockIdx.x * kWavesPerBlock + (threadIdx.x >> 5);
  const int p = wave / kDT;
  const int t = wave - p * kDT;
  const int lane = threadIdx.x & 31;
  const int half = lane >> 4;
  const int col = lane & 15;
  const int d0 = t * 16;
  const int sseg = p * kSegLen;

  v2f A[4];
  build_A(A, col, half);

  // carry into segment p: Horner over E_0..E_{p-1} with ratio alpha^SegLen
  float carry = 0.f;
  for (int q = 0; q < p; ++q)
    carry = carry * kASeg + seg_end[(size_t)q * kD + d0 + col];

  const float hscale = half ? kA8 : 1.f;  // alpha^(8*half)
  for (int c = 0; c < kChunks; ++c) {
    const int s0 = sseg + c * 16;
    v8f acc = {};
#pragma unroll
    for (int k = 0; k < 4; ++k) {
      const int r0 = 4 * k + 2 * half;
      v2f Bv;
      Bv.x = m_in[(size_t)(s0 + r0) * kD + d0 + col];
      Bv.y = m_in[(size_t)(s0 + r0 + 1) * kD + d0 + col];
      acc = __builtin_amdgcn_wmma_f32_16x16x4_f32(false, A[k], false, Bv,
                                                  (short)0, acc, false, false);
    }
    float e15 = 0.f;
#pragma unroll
    for (int r = 0; r < 8; ++r) {
      const int i = r + 8 * half;  // global row within chunk
      const float e = acc[r] + cpow(kAlpha, r + 1) * hscale * carry;
      const size_t base = (size_t)(s0 + i) * kD + d0 + col;
#pragma unroll
      for (int b = 0; b < kB; ++b) out[(size_t)b * kS * kD + base] = e;
      if (r == 7) e15 = e;
    }
    carry = bcast_upper(e15, col);
  }
}

extern "C" void kernel_launch(void* const* d_in, const int* in_sizes, int n_in,
                              void* d_out, int out_size, void* d_ws,
                              size_t ws_size, hipStream_t stream) {
  (void)in_sizes; (void)n_in; (void)out_size; (void)ws_size;
  const float* x = (const float*)d_in[0];
  float* out = (float*)d_out;
  float* m = out;                 // stage batch mean in the b=0 output slice
  float* seg_end = (float*)d_ws;  // kSeg * kD floats = 256 KB

  dim3 grid(kBlocks), block(kThreads);
  cema_partial<<<grid, block, 0, stream>>>(x, m, seg_end);
  cema_apply<<<grid, block, 0, stream>>>(m, seg_end, out);
}